// ProgressiveCuts_85426899517737
// MI455X (gfx1250) — compile-verified
//
#include <hip/hip_runtime.h>

// ProgressiveCuts on MI455X: pure data movement (0 FLOPs).
// out[b, s, 0:16000] = img[b, s*800 : s*800+16000],  b in [0,16), s in [0,180)
// Strategy: offload the entire copy to the Tensor Data Mover (TDM).
//   - Each 1-wave workgroup stages 5 overlapping windows' worth of input
//     (19200 floats = 76.8 KB) into LDS with ONE tensor_load_to_lds, then
//     emits 5 tensor_store_from_lds ops (64 KB each) from overlapping LDS
//     offsets. Adjacent groups' segments overlap: 44.2 MB is loaded for 10.24 MB of input, output is
//     written once (184.3 MB) -> ~8.4 us floor at 23.3 TB/s.
//   - Waves only build D# descriptors (SALU) and wait on TENSORcnt; no
//     VGPR data path at all.

typedef __attribute__((ext_vector_type(4))) unsigned int v4u;
typedef __attribute__((ext_vector_type(8))) int          v8i;
typedef __attribute__((ext_vector_type(4))) int          v4i;

#define WIDTH     160000
#define CROP      16000
#define STEP      800
#define S_TOTAL   180                      // (160000-16000)/800
#define GWIN      5                        // windows per workgroup
#define NG        (S_TOTAL / GWIN)         // 36 groups per batch row
#define SEG_ELEMS (CROP + (GWIN - 1) * STEP)   // 19200 floats staged in LDS
#define SEG_BYTES (SEG_ELEMS * 4)              // 76800 B  (< 320 KB WGP LDS)

// Build a 1-D contiguous-copy Tensor DMA descriptor (D# groups 0 and 1).
// Layout per CDNA5 ISA ch.7/8 (CDNA5 ISA):
//   group0: [1:0]=count=1, [63:32]=lds_addr, [120:64]=global_addr, [127:126]=type=2
//   group1: [17:16]=data_size(2 -> 4B), [79:48]=tensor_dim0, [111:80]=tensor_dim1=1,
//           [127:112]=tile_dim0, [143:128]=tile_dim1=1, [207:160]=tensor_dim0_stride
__device__ __forceinline__ void tdm_desc_1d(v4u& g0, v8i& g1,
                                            unsigned lds_byte,
                                            unsigned long long ga,
                                            unsigned nElem) {
  g0.x = 1u;                                    // count=1, user mode, no gather
  g0.y = lds_byte;                              // lds_addr (bytes)
  g0.z = (unsigned)ga;                          // global_addr[31:0]
  g0.w = ((unsigned)(ga >> 32) & 0x01FFFFFFu)   // global_addr[56:32]
         | (2u << 30);                          // type = 2 ("image")

  g1[0] = (int)(2u << 16);                      // data_size=2 (4 bytes), mask=0
  g1[1] = (int)((nElem & 0xFFFFu) << 16);       // tensor_dim0[15:0]  @ bits 63:48
  g1[2] = (int)((nElem >> 16) | (1u << 16));    // tensor_dim0[31:16] | tensor_dim1=1
  g1[3] = (int)(nElem << 16);                   // tensor_dim1[31:16]=0 | tile_dim0
  g1[4] = 1;                                    // tile_dim1=1, tile_dim2=0
  g1[5] = (int)nElem;                           // tensor_dim0_stride[31:0]
  g1[6] = 0;                                    // stride hi / dim1_stride lo
  g1[7] = 0;
}

__global__ void ProgressiveCuts_85426899517737_kernel(
    const float* __restrict__ img, float* __restrict__ out) {
  extern __shared__ float lds[];

  const unsigned gid = blockIdx.x;           // [0, B*NG)
  const unsigned b   = gid / NG;
  const unsigned grp = gid % NG;
  const unsigned s0  = grp * GWIN;

  // Low 32 bits of a generic LDS pointer are the LDS byte offset (flat->LDS).
  const unsigned lds_base = (unsigned)(unsigned long long)(const void*)lds;

  const unsigned long long src = (unsigned long long)
      (img + (unsigned long long)b * WIDTH + (unsigned long long)s0 * STEP);

  v4i z4 = {0, 0, 0, 0};
  v8i z8 = {0, 0, 0, 0, 0, 0, 0, 0};
  v4u g0; v8i g1;

  // Stage the whole overlapping segment once: HBM -> LDS (one TDM op).
  tdm_desc_1d(g0, g1, lds_base, src, SEG_ELEMS);
  __builtin_amdgcn_tensor_load_to_lds(g0, g1, z4, z4, z8, 0);
  __builtin_amdgcn_s_wait_tensorcnt(0);

  // Emit GWIN overlapping windows: LDS -> HBM (one TDM op per window).
  for (unsigned w = 0; w < GWIN; ++w) {
    const unsigned long long dst = (unsigned long long)
        (out + (unsigned long long)(b * S_TOTAL + s0 + w) * CROP);
    v4u h0; v8i h1;
    tdm_desc_1d(h0, h1, lds_base + w * (STEP * 4u), dst, CROP);
    __builtin_amdgcn_tensor_store_from_lds(h0, h1, z4, z4, z8, 0);
  }
  __builtin_amdgcn_s_wait_tensorcnt(0);
}

extern "C" void kernel_launch(void* const* d_in, const int* in_sizes, int n_in,
                              void* d_out, int out_size, void* d_ws, size_t ws_size,
                              hipStream_t stream) {
  (void)n_in; (void)d_ws; (void)ws_size; (void)out_size;
  const float* img = (const float*)d_in[0];
  float*       out = (float*)d_out;
  const int B = in_sizes[0] / WIDTH;         // 16 for the reference shapes
  dim3 grid((unsigned)(B * NG));             // 576 workgroups
  dim3 block(32);                            // exactly one wave32 -> one TDM queue
  hipLaunchKernelGGL(ProgressiveCuts_85426899517737_kernel,
                     grid, block, SEG_BYTES, stream, img, out);
}